// STN_65326452572549
// MI455X (gfx1250) — compile-verified
//
#include <hip/hip_runtime.h>
#include <math.h>

#define H_    512
#define W_    512
#define ND_   40962
#define NC_   162
#define NL_   64
#define MPAD  176
#define RADIUS_F 100.0f
#define PI_F  3.14159265358979323846f

typedef float v2f __attribute__((ext_vector_type(2)));
typedef float v4f __attribute__((ext_vector_type(4)));
typedef float v8f __attribute__((ext_vector_type(8)));

// ---- workspace layout (float offsets) ----
#define OFF_U    0               // Upad 176*64 = 11264
#define OFF_QP   11264           // Qpad (Q0) 176*64
#define OFF_PTS  22528           // pts 162*3 (pad to 512)
#define OFF_FM   23040           // fm_pad 176*176 = 30976
#define OFF_SW   54016           // sw_pad 176*176
#define OFF_CUR  84992           // cur 40962*3 = 122886 (pad)
#define OFF_DEF  207888          // def_data 40962*3
#define OFF_ACC  330784          // 4 floats

// ---------------------------------------------------------------------------
// K1: p4-normalize, U = log(clip(q)), Q0 = softmax(U), pts = unit(q . labels)
// ---------------------------------------------------------------------------
__global__ void crf_setup_kernel(const float* __restrict__ def_idxs,
                                 const float* __restrict__ labels,
                                 float* __restrict__ Upad,
                                 float* __restrict__ Qpad,
                                 float* __restrict__ pts)
{
    int r = threadIdx.x;
    if (r >= MPAD) return;
    if (r >= NC_) {                       // zero pad rows
        for (int l = 0; l < NL_; ++l) { Upad[r*NL_+l] = 0.f; Qpad[r*NL_+l] = 0.f; }
        return;
    }
    const float* x = def_idxs + r*NL_;
    float s4 = 0.f;
    for (int l = 0; l < NL_; ++l) { float v = x[l]; float v2 = v*v; s4 += v2*v2; }
    float n = fmaxf(powf(s4, 0.25f), 1e-12f);
    float t0 = 0.f, t1 = 0.f, t2 = 0.f, umax = -1e30f;
    for (int l = 0; l < NL_; ++l) {
        float qv = x[l] / n;
        const float* lb = labels + (size_t)(r*NL_+l)*3;
        t0 += qv*lb[0]; t1 += qv*lb[1]; t2 += qv*lb[2];
        float u = logf(fminf(fmaxf(qv, 1e-6f), 1.0f));
        Upad[r*NL_+l] = u;
        umax = fmaxf(umax, u);
    }
    float inv = 1.f / (sqrtf(t0*t0 + t1*t1 + t2*t2) + 1e-12f);
    pts[r*3+0] = t0*inv; pts[r*3+1] = t1*inv; pts[r*3+2] = t2*inv;
    float se = 0.f;
    for (int l = 0; l < NL_; ++l) se += expf(Upad[r*NL_+l] - umax);
    float rinv = 1.f / se;
    for (int l = 0; l < NL_; ++l) Qpad[r*NL_+l] = expf(Upad[r*NL_+l] - umax) * rinv;
}

// ---------------------------------------------------------------------------
// K2: fm_pad = filter_weights * exp(-d2/(2*gamma^2)) (zero diag), sw padded
// ---------------------------------------------------------------------------
__global__ void build_fm_sw_kernel(const float* __restrict__ fw,
                                   const float* __restrict__ sw,
                                   const float* __restrict__ pts,
                                   float* __restrict__ fm_pad,
                                   float* __restrict__ sw_pad)
{
    int i = blockIdx.x * blockDim.x + threadIdx.x;
    if (i >= MPAD*MPAD) return;
    int r = i / MPAD, c = i % MPAD;
    float f = 0.f, s = 0.f;
    if (r < NC_ && c < NC_) {
        float dx = pts[r*3+0]-pts[c*3+0];
        float dy = pts[r*3+1]-pts[c*3+1];
        float dz = pts[r*3+2]-pts[c*3+2];
        float d2 = dx*dx + dy*dy + dz*dz;
        float sf = (r == c) ? 0.f : expf(-2.0f * d2);   // 2*gamma^2 = 0.5
        f = fw[r*NC_+c] * sf;
        s = sw[r*NC_+c];
    }
    fm_pad[i] = f; sw_pad[i] = s;
}

// ---------------------------------------------------------------------------
// Wave-level f32 WMMA GEMM stage: C(176x64,LDS) = A(176xK) @ B(Kx64).
// K is a template constant so the K-loop fully unrolls into back-to-back
// v_wmma_f32_16x16x4_f32 ops. lda = K, ldb = ldc = 64.
// A-frag layout (ISA 7.12.2): lane m = lane&15, K pair = 2*(lane>>4); B mirrors.
// ---------------------------------------------------------------------------
template<int K>
__device__ __forceinline__
void gemm_stage(const float* __restrict__ A, const float* __restrict__ B,
                float* __restrict__ C, int wave, int lane)
{
    const int Nt = NL_/16;                 // 4
    const int numTiles = (MPAD/16)*Nt;     // 44
    int half = lane >> 4;                  // 0 or 1
    int l15  = lane & 15;
    for (int tile = wave; tile < numTiles; tile += 16) {   // wave-uniform branch
        int ti = tile / Nt, tj = tile % Nt;
        const float* Arow = A + (size_t)(ti*16 + l15)*K + 2*half;
        const float* Bcol = B + tj*16 + l15 + 2*half*NL_;
        v8f acc = {0.f,0.f,0.f,0.f,0.f,0.f,0.f,0.f};
        #pragma unroll
        for (int k = 0; k < K; k += 4) {
            v2f a, b;
            a.x = Arow[k];         a.y = Arow[k+1];
            b.x = Bcol[k*NL_];     b.y = Bcol[k*NL_ + NL_];
            acc = __builtin_amdgcn_wmma_f32_16x16x4_f32(false, a, false, b,
                                                        (short)0, acc, false, false);
        }
        // C/D layout: VGPR r -> (M = r + 8*half, N = l15)
        float* Crow = C + (size_t)(ti*16 + half*8)*NL_ + tj*16 + l15;
        #pragma unroll
        for (int r = 0; r < 8; ++r) Crow[(size_t)r*NL_] = acc[r];
    }
}

// ---------------------------------------------------------------------------
// K3: fully fused CRF: 5 iterations of
//       T1 = FM@Q ; T2 = SW@T1 ; T3 = T2@CM ; Q = softmax(U + T3)
// in one persistent workgroup (16 waves). Q/T buffers live in LDS (135 KB of
// the WGP's 320 KB). Ends by computing def_cont = unit(Q . labels), seeding
// the upsample buffer and zeroing the mean accumulator.
// ---------------------------------------------------------------------------
__global__ __launch_bounds__(512)
void crf_fused_kernel(const float* __restrict__ FM, const float* __restrict__ SW,
                      const float* __restrict__ CM, const float* __restrict__ Upad,
                      const float* __restrict__ Q0, const float* __restrict__ labels,
                      float* __restrict__ defcont_out, float* __restrict__ cur,
                      float* __restrict__ accum)
{
    extern __shared__ float lds[];
    float* Q  = lds;                      // 176*64
    float* TA = lds +   MPAD*NL_;         // 176*64
    float* TB = lds + 2*MPAD*NL_;         // 176*64

    int tid = threadIdx.x;
    for (int i = tid; i < MPAD*NL_; i += 512) Q[i] = Q0[i];
    __syncthreads();

    int wave = tid >> 5, lane = tid & 31;

    for (int it = 0; it < 5; ++it) {
        gemm_stage<MPAD>(FM, Q,  TA, wave, lane);   // T1 = FM @ Q   (K=176)
        __syncthreads();
        gemm_stage<MPAD>(SW, TA, TB, wave, lane);   // T2 = SW @ T1  (K=176)
        __syncthreads();
        gemm_stage<NL_>(TB, CM, TA, wave, lane);    // T3 = T2 @ CM  (K=64)
        __syncthreads();
        if (tid < NC_) {                            // Q = softmax(U + T3)
            int r = tid;
            float m = -1e30f;
            for (int l = 0; l < NL_; ++l)
                m = fmaxf(m, Upad[r*NL_+l] + TA[r*NL_+l]);
            float s = 0.f;
            for (int l = 0; l < NL_; ++l)
                s += expf(Upad[r*NL_+l] + TA[r*NL_+l] - m);
            float inv = 1.f / s;
            for (int l = 0; l < NL_; ++l)
                Q[r*NL_+l] = expf(Upad[r*NL_+l] + TA[r*NL_+l] - m) * inv;
        }
        __syncthreads();                            // pad rows of Q stay zero
    }

    if (tid < 4) accum[tid] = 0.f;
    if (tid < NC_) {                                // def_cont = unit(Q . labels)
        int r = tid;
        float t0 = 0.f, t1 = 0.f, t2 = 0.f;
        for (int l = 0; l < NL_; ++l) {
            float qv = Q[r*NL_+l];
            const float* lb = labels + (size_t)(r*NL_+l)*3;
            t0 += qv*lb[0]; t1 += qv*lb[1]; t2 += qv*lb[2];
        }
        float inv = 1.f / (sqrtf(t0*t0 + t1*t1 + t2*t2) + 1e-12f);
        t0 *= inv; t1 *= inv; t2 *= inv;
        defcont_out[r*3+0] = t0; defcont_out[r*3+1] = t1; defcont_out[r*3+2] = t2;
        cur[r*3+0] = t0; cur[r*3+1] = t1; cur[r*3+2] = t2;
    }
}

// ---------------------------------------------------------------------------
// K4: all four upsample levels in one single-block kernel. Parents always
// precede children, and levels are separated by fence + barrier.
// ---------------------------------------------------------------------------
__device__ __forceinline__
void upsample_level(float* __restrict__ cur, const int* __restrict__ idx,
                    int base, int m, int tid, int nthr)
{
    for (int i = tid; i < m; i += nthr) {
        int a = idx[2*i], b = idx[2*i+1];
        #pragma unroll
        for (int c = 0; c < 3; ++c)
            cur[(size_t)(base+i)*3+c] = 0.5f*(cur[(size_t)a*3+c] + cur[(size_t)b*3+c]);
    }
    __threadfence();
    __syncthreads();
}

__global__ __launch_bounds__(1024)
void upsample_all_kernel(float* __restrict__ cur,
                         const int* __restrict__ up3, const int* __restrict__ up4,
                         const int* __restrict__ up5, const int* __restrict__ up6)
{
    int tid = threadIdx.x;
    upsample_level(cur, up3, 162,   480,   tid, 1024);
    upsample_level(cur, up4, 642,   1920,  tid, 1024);
    upsample_level(cur, up5, 2562,  7680,  tid, 1024);
    upsample_level(cur, up6, 10242, 30720, tid, 1024);
}

// ---------------------------------------------------------------------------
// K5: unit-normalize in place + block-reduced sum for center
// ---------------------------------------------------------------------------
__global__ void unit_center_kernel(float* __restrict__ cur, float* __restrict__ accum, int N)
{
    __shared__ float sh[3][256];
    int tid = threadIdx.x;
    int i = blockIdx.x * blockDim.x + tid;
    float x = 0.f, y = 0.f, z = 0.f;
    if (i < N) {
        x = cur[3*i]; y = cur[3*i+1]; z = cur[3*i+2];
        float inv = 1.f / (sqrtf(x*x + y*y + z*z) + 1e-12f);
        x *= inv; y *= inv; z *= inv;
        cur[3*i] = x; cur[3*i+1] = y; cur[3*i+2] = z;
    }
    sh[0][tid] = x; sh[1][tid] = y; sh[2][tid] = z;
    __syncthreads();
    for (int s = 128; s > 0; s >>= 1) {
        if (tid < s) {
            sh[0][tid] += sh[0][tid+s];
            sh[1][tid] += sh[1][tid+s];
            sh[2][tid] += sh[2][tid+s];
        }
        __syncthreads();
    }
    if (tid == 0) {
        atomicAdd(accum+0, sh[0][0]);
        atomicAdd(accum+1, sh[1][0]);
        atomicAdd(accum+2, sh[2][0]);
    }
}

// ---------------------------------------------------------------------------
// K6: def_data = R*(cur - center); warps = def_data - ico_data
// ---------------------------------------------------------------------------
__global__ void finalize_kernel(const float* __restrict__ cur,
                                const float* __restrict__ accum,
                                const float* __restrict__ ico_data,
                                float* __restrict__ def_data,
                                float* __restrict__ warps_out, int N)
{
    int i = blockIdx.x * blockDim.x + threadIdx.x;
    if (i >= N) return;
    float invN = 1.0f / (float)N;
    #pragma unroll
    for (int c = 0; c < 3; ++c) {
        float d = RADIUS_F * (cur[3*i+c] - accum[c]*invN);
        def_data[3*i+c] = d;
        warps_out[3*i+c] = d - ico_data[3*i+c];
    }
}

// ---------------------------------------------------------------------------
// K7: fused lat-lon gather + bilinear sphere resample. The 512x512x64 image
// is never materialized: each point directly combines 4 bilinear corners x 3
// barycentric parents = 12 rows of moving_img (L2-resident) with folded
// coefficients. 16 lanes per point, each lane covers 4 channels (b128 loads).
// ---------------------------------------------------------------------------
__global__ void fused_resample_kernel(const float* __restrict__ def_data,
                                      const int* __restrict__ img_idxs,
                                      const float* __restrict__ img_w,
                                      const float* __restrict__ moving,
                                      float* __restrict__ warped, int N)
{
    int t = blockIdx.x * blockDim.x + threadIdx.x;
    int p = t >> 4;
    int sub = t & 15;
    if (p >= N) return;

    float vx = def_data[3*p+0] * (1.0f/RADIUS_F);
    float vy = def_data[3*p+1] * (1.0f/RADIUS_F);
    float vz = def_data[3*p+2] * (1.0f/RADIUS_F);
    vz = fminf(fmaxf(vz, -0.9999999f), 0.9999999f);
    float row = acosf(vz) * ((float)(H_-1) / PI_F);
    float col = (atan2f(vy, vx) + PI_F) * ((float)(W_-1) / (2.0f*PI_F));
    float r0f = floorf(row), c0f = floorf(col);
    float fr = row - r0f, fc = col - c0f;
    int r0 = min(max((int)r0f, 0), H_-1);
    int r1 = min(r0 + 1, H_-1);
    int c0 = min(max((int)c0f, 0), W_-1);
    int c1 = min(c0 + 1, W_-1);

    int   pix[4] = { r0*W_+c0, r0*W_+c1, r1*W_+c0, r1*W_+c1 };
    float bw[4]  = { (1.f-fr)*(1.f-fc), (1.f-fr)*fc, fr*(1.f-fc), fr*fc };

    v4f acc = {0.f, 0.f, 0.f, 0.f};
    #pragma unroll
    for (int j = 0; j < 4; ++j) {
        int base = pix[j]*3;
        #pragma unroll
        for (int k = 0; k < 3; ++k) {
            int   idx = img_idxs[base+k];
            float cw  = bw[j] * img_w[base+k];
            const v4f* rowp = (const v4f*)(moving + (size_t)idx*NL_ + sub*4);
            acc += cw * (*rowp);
        }
    }
    ((v4f*)warped)[(size_t)p*16 + sub] = acc;
}

extern "C" void kernel_launch(void* const* d_in, const int* in_sizes, int n_in,
                              void* d_out, int out_size, void* d_ws, size_t ws_size,
                              hipStream_t stream)
{
    (void)in_sizes; (void)n_in; (void)out_size; (void)ws_size;
    const float* moving   = (const float*)d_in[0];
    const float* def_idxs = (const float*)d_in[1];
    const float* labels   = (const float*)d_in[2];
    const float* ico_data = (const float*)d_in[3];
    const float* img_w    = (const float*)d_in[4];
    const float* fw       = (const float*)d_in[5];
    const float* sw       = (const float*)d_in[6];
    const float* cm       = (const float*)d_in[7];
    const int*   img_idxs = (const int*)d_in[8];
    const int*   up3      = (const int*)d_in[9];
    const int*   up4      = (const int*)d_in[10];
    const int*   up5      = (const int*)d_in[11];
    const int*   up6      = (const int*)d_in[12];

    float* ws  = (float*)d_ws;
    float* out = (float*)d_out;
    float* warped   = out;                        // 40962*64
    float* defcont  = out + (size_t)ND_*NL_;      // 162*3
    float* warpsO   = defcont + NC_*3;            // 40962*3

    float* Upad = ws + OFF_U;  float* Qpad = ws + OFF_QP; float* pts = ws + OFF_PTS;
    float* FM   = ws + OFF_FM; float* SW   = ws + OFF_SW;
    float* cur  = ws + OFF_CUR; float* defd = ws + OFF_DEF; float* acc = ws + OFF_ACC;

    crf_setup_kernel<<<1, 192, 0, stream>>>(def_idxs, labels, Upad, Qpad, pts);
    build_fm_sw_kernel<<<(MPAD*MPAD + 255)/256, 256, 0, stream>>>(fw, sw, pts, FM, SW);

    crf_fused_kernel<<<1, 512, 3*MPAD*NL_*sizeof(float), stream>>>(
        FM, SW, cm, Upad, Qpad, labels, defcont, cur, acc);

    upsample_all_kernel<<<1, 1024, 0, stream>>>(cur, up3, up4, up5, up6);

    unit_center_kernel<<<(ND_ + 255)/256, 256, 0, stream>>>(cur, acc, ND_);
    finalize_kernel<<<(ND_ + 255)/256, 256, 0, stream>>>(cur, acc, ico_data, defd, warpsO, ND_);
    fused_resample_kernel<<<(ND_*16 + 255)/256, 256, 0, stream>>>(defd, img_idxs, img_w,
                                                                  moving, warped, ND_);
}